// GRASPNET_44770739094235
// MI455X (gfx1250) — compile-verified
//
#include <hip/hip_runtime.h>
#include <math.h>

// ---- problem constants --------------------------------------------------
#define B_   16
#define G_   2000
#define EMBN 1000
#define D_   128
#define PIW_ 0.5f
#define EPS_BN 1e-5f
#define EPS_LN 1e-5f

typedef __attribute__((ext_vector_type(2))) float v2f;
typedef __attribute__((ext_vector_type(8))) float v8f;

// monotonic uint key for float atomic-max (0 is below every finite float)
__device__ __forceinline__ unsigned fkey(float f) {
  unsigned u = __float_as_uint(f);
  return (u & 0x80000000u) ? ~u : (u | 0x80000000u);
}
__device__ __forceinline__ float funkey(unsigned u) {
  unsigned v = (u & 0x80000000u) ? (u & 0x7FFFFFFFu) : ~u;
  return __uint_as_float(v);
}

template<int NT>
__device__ __forceinline__ float blockSum(float v, float* sm) {
  int t = threadIdx.x;
  sm[t] = v; __syncthreads();
  for (int s = NT >> 1; s > 0; s >>= 1) { if (t < s) sm[t] += sm[t + s]; __syncthreads(); }
  float r = sm[0]; __syncthreads();
  return r;
}
template<int NT>
__device__ __forceinline__ float blockMax(float v, float* sm) {
  int t = threadIdx.x;
  sm[t] = v; __syncthreads();
  for (int s = NT >> 1; s > 0; s >>= 1) { if (t < s) sm[t] = fmaxf(sm[t], sm[t + s]); __syncthreads(); }
  float r = sm[0]; __syncthreads();
  return r;
}

// ---- 1) per-row sums of the count matrices ------------------------------
__global__ void k_rowsum(const float* __restrict__ x1, const float* __restrict__ x2,
                         float* __restrict__ rs) {
  __shared__ float sm[256];
  int c = blockIdx.x >> 4, b = blockIdx.x & 15;
  const float* x = (c ? x2 : x1) + b * G_;
  float acc = 0.f;
  for (int i = threadIdx.x; i < G_; i += 256) acc += x[i];
  float s = blockSum<256>(acc, sm);
  if (threadIdx.x == 0) rs[blockIdx.x] = s;
}

// ---- 2) embed: normalize->index->gather->layernorm ----------------------
__global__ void k_embed(const float* __restrict__ x1, const float* __restrict__ x2,
                        const float* __restrict__ tab,
                        const float* __restrict__ lng, const float* __restrict__ lnb,
                        const float* __restrict__ rs, float* __restrict__ e) {
  __shared__ float sm[128];
  int t = blockIdx.x;                  // 0 .. 2*B*G-1
  int c = t / (B_ * G_);
  int r = t - c * B_ * G_;
  int b = r / G_, g = r - b * G_;
  const float* x = c ? x2 : x1;
  float xv = x[b * G_ + g];
  int idx = (int)(xv / rs[c * 16 + b] * (float)(EMBN - 1));
  idx = min(max(idx, 0), EMBN - 1);
  int d = threadIdx.x;
  float v = tab[idx * D_ + d];
  float m = blockSum<128>(v, sm) * (1.f / D_);
  float dv = v - m;
  float var = blockSum<128>(dv * dv, sm) * (1.f / D_);
  e[(size_t)t * D_ + d] = lng[d] * dv * rsqrtf(var + EPS_LN) + lnb[d];
}

// ---- 3) ws = softmax(pool_w); u = Wv @ ws -------------------------------
__global__ void k_poolprep(const float* __restrict__ pool_w, const float* __restrict__ Wv,
                           float* __restrict__ u) {
  __shared__ float sm[128];
  __shared__ float w[128];
  int t = threadIdx.x;
  float pv = pool_w[t];
  float mx = blockMax<128>(pv, sm);
  float ev = __expf(pv - mx);
  float s = blockSum<128>(ev, sm);
  w[t] = ev / s;
  __syncthreads();
  float acc = 0.f;
  for (int n = 0; n < 128; ++n) acc += Wv[t * 128 + n] * w[n];
  u[t] = acc;
}

// ---- 4) vw[row] = e[row,:] . u   (one wave32 per row) -------------------
__global__ void k_vw(const float* __restrict__ e, const float* __restrict__ u,
                     float* __restrict__ vw) {
  int row = (blockIdx.x << 3) + (threadIdx.x >> 5);
  int l = threadIdx.x & 31;
  const float* rp = e + (size_t)row * D_;
  float acc = 0.f;
  for (int i = l; i < D_; i += 32) acc += rp[i] * u[i];
  for (int o = 16; o; o >>= 1) acc += __shfl_down(acc, o, 32);
  if (l == 0) vw[row] = acc;
}

// ---- 5) projections eq = e@Wq, ek = e@Wk via V_WMMA_F32_16X16X4_F32 -----
// one wave per 16x16 output tile; K=128 in 32 steps of 4
__global__ void k_proj(const float* __restrict__ e,
                       const float* __restrict__ Wq, const float* __restrict__ Wk,
                       float* __restrict__ eq, float* __restrict__ ek) {
  int tile = (blockIdx.x << 3) + (threadIdx.x >> 5);  // 0..63999
  int wsel = tile / 32000;                            // 0 -> Wq, 1 -> Wk
  int rem = tile - wsel * 32000;
  int rowtile = rem >> 3;                             // 0..3999 (16-row tiles over 2*B*G)
  int ntile = rem & 7;                                // 8 column tiles of 16
  const float* W = wsel ? Wk : Wq;
  const float* src = e + (size_t)rowtile * 16 * D_;
  float* dst = (wsel ? ek : eq) + (size_t)rowtile * 16 * D_;
  int l = threadIdx.x & 31;
  int m = l & 15, kh = l >> 4;                        // fragment row/col, K half
  int n0 = ntile * 16;
  v8f c = {};
  for (int k0 = 0; k0 < D_; k0 += 4) {
    v2f a, bfr;
    a.x = src[m * D_ + k0 + 2 * kh];
    a.y = src[m * D_ + k0 + 2 * kh + 1];
    bfr.x = W[(k0 + 2 * kh) * D_ + n0 + m];
    bfr.y = W[(k0 + 2 * kh + 1) * D_ + n0 + m];
    c = __builtin_amdgcn_wmma_f32_16x16x4_f32(false, a, false, bfr, (short)0, c, false, false);
  }
  for (int r2 = 0; r2 < 8; ++r2)
    dst[(r2 + 8 * kh) * D_ + n0 + m] = c[r2];
}

// ---- 6) scores S = q @ k^T / sqrt(D) ------------------------------------
// macro-tiled: each wave owns 80 rows (5 stacked 16x16 tiles) x 16 cols,
// so the B (key) fragment is loaded once per K-step and reused by 5 WMMAs.
__global__ void k_scores(const float* __restrict__ eq, const float* __restrict__ ek,
                         float* __restrict__ aw, unsigned* __restrict__ gmax) {
  int tile = (blockIdx.x << 3) + (threadIdx.x >> 5);  // 0..99999
  int a = tile / (16 * 25 * 125);
  int rem = tile - a * (16 * 25 * 125);
  int b = rem / (25 * 125);
  int rem2 = rem - b * (25 * 125);
  int mt = rem2 / 125;                                // macro row tile (80 rows)
  int nt = rem2 - mt * 125;                           // column tile (16 cols)
  const float* qb = eq + (((size_t)a * B_ + b) * G_ + (size_t)mt * 80) * D_;
  const float* kb = ek + (((size_t)(1 - a) * B_ + b) * G_ + (size_t)nt * 16) * D_;
  int l = threadIdx.x & 31;
  int m = l & 15, kh = l >> 4;
  v8f c[5];
#pragma unroll
  for (int t5 = 0; t5 < 5; ++t5) c[t5] = (v8f){};
  for (int k0 = 0; k0 < D_; k0 += 4) {
    int ko = k0 + 2 * kh;
    v2f bv;
    bv.x = kb[m * D_ + ko];               // B[k][n] = kp[n][k]
    bv.y = kb[m * D_ + ko + 1];
#pragma unroll
    for (int t5 = 0; t5 < 5; ++t5) {
      v2f av;
      av.x = qb[(m + 16 * t5) * D_ + ko];
      av.y = qb[(m + 16 * t5) * D_ + ko + 1];
      c[t5] = __builtin_amdgcn_wmma_f32_16x16x4_f32(false, av, false, bv, (short)0, c[t5],
                                                    false, false);
    }
  }
  const float sc = 0.08838834764831845f;  // 1/sqrt(128)
  float* ob = aw + (size_t)a * ((size_t)B_ * G_ * G_) + ((size_t)b * G_ + mt * 80) * G_ + nt * 16;
  float lm = -3.4e38f;
#pragma unroll
  for (int t5 = 0; t5 < 5; ++t5) {
#pragma unroll
    for (int r2 = 0; r2 < 8; ++r2) {
      float v = c[t5][r2] * sc;
      ob[(size_t)(16 * t5 + r2 + 8 * kh) * G_ + m] = v;
      lm = fmaxf(lm, v);
    }
  }
  for (int o = 16; o; o >>= 1) lm = fmaxf(lm, __shfl_down(lm, o, 32));
  if (l == 0) atomicMax(&gmax[a], fkey(lm));
}

// ---- 7) per-row softmax (+diag bias) + entropy + fused pooling ----------
__global__ void k_softmax(float* __restrict__ aw, const float* __restrict__ vw,
                          const unsigned* __restrict__ gmax,
                          float* __restrict__ ent_a, float* __restrict__ pout) {
  __shared__ float sm[256];
  int t = blockIdx.x;                 // 0..63999 rows
  int a = t / (B_ * G_);
  int r = t - a * B_ * G_;
  int b = r / G_, g = r - b * G_;
  float* row = aw + (size_t)a * ((size_t)B_ * G_ * G_) + ((size_t)b * G_ + g) * G_;
  const float* vwr = vw + ((size_t)(1 - a) * B_ + b) * G_;
  float bias = PIW_ * funkey(gmax[a]);
  int tid = threadIdx.x;
  float vals[8];
  float lmax = -3.4e38f;
  for (int i = 0; i < 8; ++i) {
    int h = tid + (i << 8);
    float v = -3.4e38f;
    if (h < G_) { v = row[h]; if (h == g) v += bias; }
    vals[i] = v;
    lmax = fmaxf(lmax, v);
  }
  float mx = blockMax<256>(lmax, sm);
  float lsum = 0.f;
  for (int i = 0; i < 8; ++i) {
    int h = tid + (i << 8);
    float ev = (h < G_) ? __expf(vals[i] - mx) : 0.f;
    vals[i] = ev;
    lsum += ev;
  }
  float s = blockSum<256>(lsum, sm);
  float inv = 1.f / s;
  float lent = 0.f, lpool = 0.f;
  for (int i = 0; i < 8; ++i) {
    int h = tid + (i << 8);
    if (h < G_) {
      float p = vals[i] * inv;
      row[h] = p;
      lent += p * __logf(p + 1e-10f);
      lpool += p * vwr[h];
    }
  }
  float ent = blockSum<256>(lent, sm);
  float pool = blockSum<256>(lpool, sm);
  if (tid == 0) {
    atomicAdd(&ent_a[a], ent);
    pout[(size_t)a * B_ * G_ + (size_t)b * G_ + g] = pool;
  }
}

// ---- 8) fused Linear -> BatchNorm1d(train) -> ReLU ----------------------
__global__ void k_lbr(const float* __restrict__ x, const float* __restrict__ W,
                      const float* __restrict__ bias, const float* __restrict__ gamma,
                      const float* __restrict__ beta, float* __restrict__ out,
                      int din, int dout) {
  __shared__ float sm[128];
  __shared__ float yv[16];
  __shared__ float stats[2];
  int j = blockIdx.x, tid = threadIdx.x;
  for (int b = 0; b < 16; ++b) {
    float acc = 0.f;
    for (int i = tid; i < din; i += 128) acc += x[(size_t)b * din + i] * W[(size_t)i * dout + j];
    float s = blockSum<128>(acc, sm);
    if (tid == 0) yv[b] = s + bias[j];
    __syncthreads();
  }
  if (tid == 0) {
    float m = 0.f;
    for (int b = 0; b < 16; ++b) m += yv[b];
    m *= (1.f / 16.f);
    float v = 0.f;
    for (int b = 0; b < 16; ++b) { float d = yv[b] - m; v += d * d; }
    v *= (1.f / 16.f);
    stats[0] = m; stats[1] = rsqrtf(v + EPS_BN);
  }
  __syncthreads();
  if (tid < 16) {
    float o = gamma[j] * (yv[tid] - stats[0]) * stats[1] + beta[j];
    out[(size_t)tid * dout + j] = fmaxf(o, 0.f);
  }
}

// ---- 9) entropy of softmax(h, axis=1) -----------------------------------
__global__ void k_hent(const float* __restrict__ h1, const float* __restrict__ h2,
                       float* __restrict__ entc) {
  __shared__ float sm[128];
  int which = blockIdx.x >> 4, b = blockIdx.x & 15;
  const float* row = (which ? h2 : h1) + b * 128;
  int t = threadIdx.x;
  float v = row[t];
  float mx = blockMax<128>(v, sm);
  float ev = __expf(v - mx);
  float s = blockSum<128>(ev, sm);
  float p = ev / s;
  float ent = blockSum<128>(p * __logf(p + 1e-10f), sm);
  if (t == 0) atomicAdd(&entc[which], ent);
}

// ---- 10) write scalar outputs -------------------------------------------
__global__ void k_final(const float* __restrict__ ent_a, const float* __restrict__ entc,
                        float* __restrict__ el) {
  el[0] = -ent_a[0] * (1.f / (float)(B_ * G_));
  el[1] = -ent_a[1] * (1.f / (float)(B_ * G_));
  el[2] = -entc[0] * (1.f / 16.f);
  el[3] = -entc[1] * (1.f / 16.f);
}

extern "C" void kernel_launch(void* const* d_in, const int* in_sizes, int n_in,
                              void* d_out, int out_size, void* d_ws, size_t ws_size,
                              hipStream_t stream) {
  (void)in_sizes; (void)n_in; (void)out_size; (void)ws_size;
  const float* x1    = (const float*)d_in[0];
  const float* x2    = (const float*)d_in[1];
  const float* tab   = (const float*)d_in[2];
  const float* lng   = (const float*)d_in[3];
  const float* lnb   = (const float*)d_in[4];
  const float* Wq    = (const float*)d_in[5];
  const float* Wk    = (const float*)d_in[6];
  const float* Wv    = (const float*)d_in[7];
  const float* poolw = (const float*)d_in[8];
  const float* eW[3] = {(const float*)d_in[9],  (const float*)d_in[13], (const float*)d_in[17]};
  const float* eb[3] = {(const float*)d_in[10], (const float*)d_in[14], (const float*)d_in[18]};
  const float* eg[3] = {(const float*)d_in[11], (const float*)d_in[15], (const float*)d_in[19]};
  const float* ee[3] = {(const float*)d_in[12], (const float*)d_in[16], (const float*)d_in[20]};
  const float* pW[2] = {(const float*)d_in[21], (const float*)d_in[25]};
  const float* pb[2] = {(const float*)d_in[22], (const float*)d_in[26]};
  const float* pg[2] = {(const float*)d_in[23], (const float*)d_in[27]};
  const float* pe[2] = {(const float*)d_in[24], (const float*)d_in[28]};

  // workspace layout (floats)
  float* ws = (float*)d_ws;
  size_t off = 0;
  float* e  = ws + off; off += (size_t)2 * B_ * G_ * D_;
  float* eq = ws + off; off += (size_t)2 * B_ * G_ * D_;
  float* ek = ws + off; off += (size_t)2 * B_ * G_ * D_;
  float* vw = ws + off; off += (size_t)2 * B_ * G_;
  float* pp = ws + off; off += (size_t)2 * B_ * G_;
  float* u  = ws + off; off += 128;
  float* rs = ws + off; off += 32;
  float* t0 = ws + off; off += 16 * 512;
  float* t1 = ws + off; off += 16 * 256;
  float* t2 = ws + off; off += 16 * 128;
  unsigned* gmax = (unsigned*)(ws + off); off += 2;
  float* ent_a = ws + off; off += 2;
  float* entc  = ws + off; off += 2;

  // output layout
  float* out = (float*)d_out;
  float* h1o = out;                 // [16,128]
  float* h2o = out + 2048;          // [16,128]
  float* z1o = out + 4096;          // [16,64]
  float* z2o = out + 5120;          // [16,64]
  float* aw  = out + 6144;          // aw1 then aw2, 2*16*2000*2000
  float* el  = out + 6144 + (size_t)2 * B_ * G_ * G_;  // 4 scalars

  // zero the accumulators (gmax keys + entropy sums, contiguous)
  hipMemsetAsync(gmax, 0, 6 * sizeof(unsigned), stream);

  k_rowsum<<<32, 256, 0, stream>>>(x1, x2, rs);
  k_embed<<<2 * B_ * G_, 128, 0, stream>>>(x1, x2, tab, lng, lnb, rs, e);
  k_poolprep<<<1, 128, 0, stream>>>(poolw, Wv, u);
  k_vw<<<(2 * B_ * G_) / 8, 256, 0, stream>>>(e, u, vw);
  k_proj<<<8000, 256, 0, stream>>>(e, Wq, Wk, eq, ek);
  k_scores<<<12500, 256, 0, stream>>>(eq, ek, aw, gmax);
  k_softmax<<<2 * B_ * G_, 256, 0, stream>>>(aw, vw, gmax, ent_a, pp);

  // encoder + projection stacks for both branches
  for (int which = 0; which < 2; ++which) {
    const float* pin = pp + (size_t)which * B_ * G_;
    float* ho = which ? h2o : h1o;
    float* zo = which ? z2o : z1o;
    k_lbr<<<512, 128, 0, stream>>>(pin, eW[0], eb[0], eg[0], ee[0], t0, 2000, 512);
    k_lbr<<<256, 128, 0, stream>>>(t0,  eW[1], eb[1], eg[1], ee[1], t1, 512, 256);
    k_lbr<<<128, 128, 0, stream>>>(t1,  eW[2], eb[2], eg[2], ee[2], ho, 256, 128);
    k_lbr<<<128, 128, 0, stream>>>(ho,  pW[0], pb[0], pg[0], pe[0], t2, 128, 128);
    k_lbr<<<64,  128, 0, stream>>>(t2,  pW[1], pb[1], pg[1], pe[1], zo, 128, 64);
  }

  k_hent<<<32, 128, 0, stream>>>(h1o, h2o, entc);
  k_final<<<1, 1, 0, stream>>>(ent_a, entc, el);
}